// InverseWaveletBlock2D_46162308498070
// MI455X (gfx1250) — compile-verified
//
#include <hip/hip_runtime.h>
#include <hip/hip_bf16.h>

typedef __attribute__((ext_vector_type(2))) float v2f;
typedef __attribute__((ext_vector_type(8))) float v8f;

#define TILE_IN   30              // input rows/cols reconstructed per block
#define TILE_OUT  60              // = 2*TILE_IN output rows/cols per block
#define IN_STRIDE 38              // 32 valid + pad (even -> 8B-aligned v2f loads, conflict-free)
#define PQ_STRIDE 65              // 64 valid cols + 1 pad

#ifndef __has_builtin
#define __has_builtin(x) 0
#endif
#if __has_builtin(__builtin_amdgcn_global_load_async_to_lds_b32)
#define USE_ASYNC_LDS 1
#else
#define USE_ASYNC_LDS 0
#endif

#if USE_ASYNC_LDS
typedef __attribute__((address_space(1))) int g_int_t;   // printed as "__device__ int" by clang
typedef __attribute__((address_space(3))) int l_int_t;   // printed as "__shared__ int"
#endif

__device__ __forceinline__ int reflect_idx(int i, int n) {
    if (i < 0)  i = -i;             // -1 -> 1  (matches jnp reflect, PAD=1)
    if (i >= n) i = 2 * n - 2 - i;  //  n -> n-2
    if (i < 0)  i = 0;              // safety clamp for over-computed halo
    if (i >= n) i = n - 1;
    return i;
}

__global__ __launch_bounds__(256, 2)
void iwt2d_fused_wmma(const float* __restrict__ ss, const float* __restrict__ sd,
                      const float* __restrict__ ds, const float* __restrict__ dd,
                      const float* __restrict__ h6, const float* __restrict__ g6,
                      float* __restrict__ out, int Hin, int Win)
{
    // LDS: 4 input tiles (32 rows x IN_STRIDE), P/Q intermediates (36 x PQ_STRIDE), filters
    __shared__ __align__(16) float s_in[4][32 * IN_STRIDE];
    __shared__ __align__(16) float s_P[36 * PQ_STRIDE];
    __shared__ __align__(16) float s_Q[36 * PQ_STRIDE];
    __shared__ float s_f[12];

    const int Hout = 2 * Hin, Wout = 2 * Win;
    const int tid  = threadIdx.x;
    const int wave = tid >> 5;
    const int lane = tid & 31;
    const int rIdx = lane & 15;            // lane's M (A-role) or N (B-role) index
    const int half = (lane < 16) ? 0 : 2;  // K offset of VGPR component 0

    const int r0 = blockIdx.y * TILE_IN;   // input-tile origin (virtual halo row = r0-1)
    const int c0 = blockIdx.x * TILE_IN;
    const int img = blockIdx.z;            // flattened (b, c)
    const size_t in_off  = (size_t)img * Hin * Win;
    const size_t out_off = (size_t)img * Hout * Wout;

    // ---- filters to LDS ----
    if (tid < 12) s_f[tid] = (tid < 6) ? h6[tid] : g6[tid - 6];

    // ---- zero K-pad regions (padded taps must multiply finite zeros) ----
    for (int i = tid; i < 32 * 6; i += 256) {          // input cols 32..37
        int r = i / 6, c = 32 + i % 6;
        s_in[0][r * IN_STRIDE + c] = 0.f; s_in[1][r * IN_STRIDE + c] = 0.f;
        s_in[2][r * IN_STRIDE + c] = 0.f; s_in[3][r * IN_STRIDE + c] = 0.f;
    }
    for (int i = tid; i < 4 * PQ_STRIDE; i += 256) {   // P/Q rows 32..35
        s_P[32 * PQ_STRIDE + i] = 0.f;
        s_Q[32 * PQ_STRIDE + i] = 0.f;
    }

    // ---- load 32x32 input tiles with reflect mapping (coalesced in c) ----
    for (int i = tid; i < 32 * 32; i += 256) {
        int r = i >> 5, c = i & 31;
        int gr = reflect_idx(r0 - 1 + r, Hin);
        int gc = reflect_idx(c0 - 1 + c, Win);
        size_t gi = in_off + (size_t)gr * Win + gc;
        int li = r * IN_STRIDE + c;
#if USE_ASYNC_LDS
        // Direct global->LDS gather copy (ASYNCcnt-tracked), avoids VGPR staging.
        __builtin_amdgcn_global_load_async_to_lds_b32(
            (g_int_t*)(uintptr_t)(ss + gi), (l_int_t*)(uintptr_t)&s_in[0][li], 0, 0);
        __builtin_amdgcn_global_load_async_to_lds_b32(
            (g_int_t*)(uintptr_t)(sd + gi), (l_int_t*)(uintptr_t)&s_in[1][li], 0, 0);
        __builtin_amdgcn_global_load_async_to_lds_b32(
            (g_int_t*)(uintptr_t)(ds + gi), (l_int_t*)(uintptr_t)&s_in[2][li], 0, 0);
        __builtin_amdgcn_global_load_async_to_lds_b32(
            (g_int_t*)(uintptr_t)(dd + gi), (l_int_t*)(uintptr_t)&s_in[3][li], 0, 0);
#else
        s_in[0][li] = ss[gi]; s_in[1][li] = sd[gi];
        s_in[2][li] = ds[gi]; s_in[3][li] = dd[gi];
#endif
    }
#if USE_ASYNC_LDS
#if __has_builtin(__builtin_amdgcn_s_wait_asynccnt)
    __builtin_amdgcn_s_wait_asynccnt(0);
#else
    asm volatile("s_wait_asynccnt 0" ::: "memory");
#endif
#endif
    __syncthreads();

    // ---- build filter WMMA operands: F[r][k] = f[r&1][k - (r>>1)], tile-invariant.
    //      Same registers serve B-role (stage 1) and A-role (stage 2).
    //      Polyphase: f[p][j] = tap[4 - 2j + p]  (tap = h or g)
    v2f Fh[3], Fg[3];
    {
        const int pp = rIdx & 1, nn = rIdx >> 1;
#pragma unroll
        for (int cch = 0; cch < 3; ++cch) {
#pragma unroll
            for (int e = 0; e < 2; ++e) {
                const int k = 4 * cch + half + e;
                const int j = k - nn;
                const bool ok = (j >= 0) && (j < 3);
                const int jc = ok ? j : 0;
                Fh[cch][e] = ok ? s_f[4 - 2 * jc + pp]     : 0.f;
                Fg[cch][e] = ok ? s_f[6 + 4 - 2 * jc + pp] : 0.f;
            }
        }
    }

    // ================= Stage 1: W-synthesis ==================
    // P = ss (*)w Fh + ds (*)w Fg ; Q = sd (*)w Fh + dd (*)w Fg
    // 16 accumulator tiles (P:8, Q:8); each wave owns 2.
#pragma unroll
    for (int tt = 0; tt < 2; ++tt) {
        const int t  = wave * 2 + tt;
        const int q  = t >> 3;               // 0 -> P, 1 -> Q
        const int mi = (t >> 2) & 1;         // h 16-tile
        const int ni = t & 3;                // wo 16-tile
        const float* A0 = s_in[q ? 1 : 0];   // paired with Fh
        const float* A1 = s_in[q ? 3 : 2];   // paired with Fg
        float* Pout = q ? s_Q : s_P;

        const int hrow = 16 * mi + rIdx;     // A-role: lane holds row M
        v8f acc = {};
#pragma unroll
        for (int cch = 0; cch < 3; ++cch) {
            const int col = 8 * ni + 4 * cch + half;   // even -> 8B aligned
            v2f a0 = *(const v2f*)&A0[hrow * IN_STRIDE + col];
            acc = __builtin_amdgcn_wmma_f32_16x16x4_f32(
                      false, a0, false, Fh[cch], (short)0, acc, false, false);
            v2f a1 = *(const v2f*)&A1[hrow * IN_STRIDE + col];
            acc = __builtin_amdgcn_wmma_f32_16x16x4_f32(
                      false, a1, false, Fg[cch], (short)0, acc, false, false);
        }
        // D layout: VGPR r -> M = r + (lane<16 ? 0 : 8), N = rIdx
        const int baseM = 16 * mi + ((lane < 16) ? 0 : 8);
        const int colo  = 16 * ni + rIdx;
#pragma unroll
        for (int r = 0; r < 8; ++r)
            Pout[(baseM + r) * PQ_STRIDE + colo] = acc[r];
    }
    __syncthreads();

    // ================= Stage 2: H-synthesis ==================
    // out = Fh (*)h P + Fg (*)h Q ; 16 output tiles, 2 per wave.
#pragma unroll
    for (int tt = 0; tt < 2; ++tt) {
        const int t  = wave * 2 + tt;
        const int mi = t >> 2;               // ho 16-tile (0..3)
        const int ni = t & 3;                // wo 16-tile (0..3)
        const int colo = 16 * ni + rIdx;     // B-role: lane holds column N
        v8f acc = {};
#pragma unroll
        for (int cch = 0; cch < 3; ++cch) {
            const int krow = 8 * mi + 4 * cch + half;   // P row = K (component +1 -> next row)
            v2f bP, bQ;
            bP[0] = s_P[krow * PQ_STRIDE + colo];
            bP[1] = s_P[(krow + 1) * PQ_STRIDE + colo];
            acc = __builtin_amdgcn_wmma_f32_16x16x4_f32(
                      false, Fh[cch], false, bP, (short)0, acc, false, false);
            bQ[0] = s_Q[krow * PQ_STRIDE + colo];
            bQ[1] = s_Q[(krow + 1) * PQ_STRIDE + colo];
            acc = __builtin_amdgcn_wmma_f32_16x16x4_f32(
                      false, Fg[cch], false, bQ, (short)0, acc, false, false);
        }
        // store valid 60x60 window, clipped to the image
        const int baseM = 16 * mi + ((lane < 16) ? 0 : 8);
        const int wo = 16 * ni + rIdx;
        const int gc = 2 * c0 + wo;
#pragma unroll
        for (int r = 0; r < 8; ++r) {
            const int ho = baseM + r;
            const int gr = 2 * r0 + ho;
            if (ho < TILE_OUT && wo < TILE_OUT && gr < Hout && gc < Wout)
                out[out_off + (size_t)gr * Wout + gc] = acc[r];
        }
    }
}

extern "C" void kernel_launch(void* const* d_in, const int* in_sizes, int n_in,
                              void* d_out, int out_size, void* d_ws, size_t ws_size,
                              hipStream_t stream) {
    const float* ss = (const float*)d_in[0];
    const float* sd = (const float*)d_in[1];
    const float* ds = (const float*)d_in[2];
    const float* dd = (const float*)d_in[3];
    const float* h6 = (const float*)d_in[4];
    const float* g6 = (const float*)d_in[5];
    float* out = (float*)d_out;

    const int Hin = 256, Win = 256;
    const int images = in_sizes[0] / (Hin * Win);   // B*C = 256
    dim3 grid((2 * Win + TILE_OUT - 1) / TILE_OUT,
              (2 * Hin + TILE_OUT - 1) / TILE_OUT,
              images);
    iwt2d_fused_wmma<<<grid, 256, 0, stream>>>(ss, sd, ds, dd, h6, g6, out, Hin, Win);
}